// HeteroGNN_59468117180546
// MI455X (gfx1250) — compile-verified
//
#include <hip/hip_runtime.h>
#include <hip/hip_bf16.h>

typedef __attribute__((ext_vector_type(2))) float v2f;
typedef __attribute__((ext_vector_type(8))) float v8f;

#define NN 50000
#define EE 500000
#define PAIR_STRIDE 544               // 512 + 32 pad: half-waves hit disjoint LDS bank windows
#define WP_FLOATS (32 * PAIR_STRIDE)  // 17408 floats = 68KB

__device__ __forceinline__ v8f wmma_f32(v2f a, v2f b, v8f c) {
  // V_WMMA_F32_16X16X4_F32: D = A(16x4) * B(4x16) + C(16x16), all fp32
  return __builtin_amdgcn_wmma_f32_16x16x4_f32(false, a, false, b, (short)0, c, false, false);
}

__device__ __forceinline__ unsigned fenc(float f) {
  unsigned u = __float_as_uint(f);
  return (u & 0x80000000u) ? ~u : (u | 0x80000000u);
}
__device__ __forceinline__ float fdec(unsigned u) {
  return (u & 0x80000000u) ? __uint_as_float(u ^ 0x80000000u) : __uint_as_float(~u);
}

// Repack W[64,256] -> Wp[pair][n][2] with padded pair stride (one b64 per B fragment).
__global__ void repack_w(const float* __restrict__ W, float* __restrict__ Wp) {
  int i = blockIdx.x * 256 + threadIdx.x;  // 16384 elements
  int k = i >> 8, n = i & 255;
  Wp[(k >> 1) * PAIR_STRIDE + n * 2 + (k & 1)] = W[i];
}

// ---------------- GEMM: C[M,256] = A[M,64] * W[64,256] (fp32 WMMA, B in LDS) ----------------
__global__ __launch_bounds__(256) void gemm_64x256(const float* __restrict__ A,
                                                   const float* __restrict__ Wp,
                                                   float* __restrict__ C, int M) {
  __shared__ float Bs[WP_FLOATS];  // 68KB of 320KB/WGP
  {
    const float4* s4 = (const float4*)Wp;
    float4* d4 = (float4*)Bs;
    for (int i = threadIdx.x; i < WP_FLOATS / 4; i += 256) d4[i] = s4[i];
  }
  __syncthreads();

  int wid = threadIdx.x >> 5;
  int lane = threadIdx.x & 31;
  int row = lane & 15;
  int kh = (lane >> 4) * 2;  // K half: 0 or 2
  int nl = lane & 15;

  // block covers 4 m-tiles x 4 n-groups = 16 wave-tasks over 8 waves
  for (int t = wid; t < 16; t += 8) {
    int m_tile = blockIdx.x * 4 + (t >> 2);
    int n_grp = t & 3;
    if (m_tile * 16 >= M) break;  // wave-uniform
    const float* Ar = A + (m_tile * 16 + row) * 64;
    int nc2 = (n_grp * 64 + nl) * 2;
    v8f acc0 = {}, acc1 = {}, acc2 = {}, acc3 = {};
    #pragma unroll 4
    for (int kk = 0; kk < 64; kk += 4) {
      v2f a; a.x = Ar[kk + kh]; a.y = Ar[kk + kh + 1];
      const float* bp = Bs + ((kk + kh) >> 1) * PAIR_STRIDE + nc2;
      v2f b;
      b = *(const v2f*)(bp +  0); acc0 = wmma_f32(a, b, acc0);
      b = *(const v2f*)(bp + 32); acc1 = wmma_f32(a, b, acc1);
      b = *(const v2f*)(bp + 64); acc2 = wmma_f32(a, b, acc2);
      b = *(const v2f*)(bp + 96); acc3 = wmma_f32(a, b, acc3);
    }
    int mrow = m_tile * 16 + ((lane >> 4) ? 8 : 0);
    int cc = n_grp * 64 + nl;
    #pragma unroll
    for (int r = 0; r < 8; ++r) {
      float* Cr = C + (mrow + r) * 256 + cc;
      Cr[ 0] = acc0[r]; Cr[16] = acc1[r]; Cr[32] = acc2[r]; Cr[48] = acc3[r];
    }
  }
}

// ------- fused out: C = relu(Agg)@P + X@S + (pb+sb), all [M,64]x[64,64] WMMA -------
__global__ __launch_bounds__(256) void out_proj(const float* __restrict__ Agg,
                                                const float* __restrict__ X,
                                                const float* __restrict__ P,
                                                const float* __restrict__ S,
                                                const float* __restrict__ pb,
                                                const float* __restrict__ sb,
                                                float* __restrict__ C, int M) {
  int gwid = blockIdx.x * 8 + (threadIdx.x >> 5);
  int lane = threadIdx.x & 31;
  if (gwid * 16 >= M) return;
  int row = lane & 15;
  int kh = (lane >> 4) * 2;
  int nc = lane & 15;
  v8f acc0 = {}, acc1 = {}, acc2 = {}, acc3 = {};
  const float* Ar = Agg + (gwid * 16 + row) * 64;
  #pragma unroll 4
  for (int kk = 0; kk < 64; kk += 4) {
    v2f a; a.x = fmaxf(Ar[kk + kh], 0.f); a.y = fmaxf(Ar[kk + kh + 1], 0.f);
    const float* B0 = P + (kk + kh) * 64;
    const float* B1 = B0 + 64;
    v2f b;
    b.x = B0[nc +  0]; b.y = B1[nc +  0]; acc0 = wmma_f32(a, b, acc0);
    b.x = B0[nc + 16]; b.y = B1[nc + 16]; acc1 = wmma_f32(a, b, acc1);
    b.x = B0[nc + 32]; b.y = B1[nc + 32]; acc2 = wmma_f32(a, b, acc2);
    b.x = B0[nc + 48]; b.y = B1[nc + 48]; acc3 = wmma_f32(a, b, acc3);
  }
  const float* Xr = X + (gwid * 16 + row) * 64;
  #pragma unroll 4
  for (int kk = 0; kk < 64; kk += 4) {
    v2f a; a.x = Xr[kk + kh]; a.y = Xr[kk + kh + 1];
    const float* B0 = S + (kk + kh) * 64;
    const float* B1 = B0 + 64;
    v2f b;
    b.x = B0[nc +  0]; b.y = B1[nc +  0]; acc0 = wmma_f32(a, b, acc0);
    b.x = B0[nc + 16]; b.y = B1[nc + 16]; acc1 = wmma_f32(a, b, acc1);
    b.x = B0[nc + 32]; b.y = B1[nc + 32]; acc2 = wmma_f32(a, b, acc2);
    b.x = B0[nc + 48]; b.y = B1[nc + 48]; acc3 = wmma_f32(a, b, acc3);
  }
  float bi0 = pb[nc +  0] + sb[nc +  0];
  float bi1 = pb[nc + 16] + sb[nc + 16];
  float bi2 = pb[nc + 32] + sb[nc + 32];
  float bi3 = pb[nc + 48] + sb[nc + 48];
  int mrow = gwid * 16 + ((lane >> 4) ? 8 : 0);
  #pragma unroll
  for (int r = 0; r < 8; ++r) {
    float* Cr = C + (mrow + r) * 64 + nc;
    Cr[ 0] = acc0[r] + bi0; Cr[16] = acc1[r] + bi1;
    Cr[32] = acc2[r] + bi2; Cr[48] = acc3[r] + bi3;
  }
}

// -------- al[n,h] = dot(H[n, h*64:(h+1)*64], avec[h,:]) ; wave per node ----------
__global__ __launch_bounds__(256) void attn_logits(const float* __restrict__ H,
                                                   const float* __restrict__ avec,
                                                   float* __restrict__ al, int n) {
  int node = blockIdx.x * 8 + (threadIdx.x >> 5);
  int lane = threadIdx.x & 31;
  if (node >= n) return;
  int h = lane >> 3;
  int c0 = (lane & 7) * 8;
  const float* Hp = H + node * 256 + h * 64 + c0;
  const float* ap = avec + h * 64 + c0;
  float s = 0.f;
  #pragma unroll
  for (int j = 0; j < 8; ++j) s += Hp[j] * ap[j];
  s += __shfl_down(s, 4, 8);
  s += __shfl_down(s, 2, 8);
  s += __shfl_down(s, 1, 8);
  if ((lane & 7) == 0) al[node * 4 + h] = s;
}

__global__ void init_agg(float* __restrict__ agg, const float* __restrict__ b1,
                         const float* __restrict__ b2, int total) {
  int i = blockIdx.x * blockDim.x + threadIdx.x;
  if (i < total) agg[i] = b1[i & 63] + b2[i & 63];
}

__global__ void init_md(unsigned* __restrict__ menc, float* __restrict__ denom, int total) {
  int i = blockIdx.x * blockDim.x + threadIdx.x;
  if (i < total) { menc[i] = 0u; denom[i] = 0.f; }
}

// pass A: e = leaky_relu(al_s[src]+al_d[dst]); store; segment max (uint-encoded)
__global__ void edge_pass_a(const int* __restrict__ src, const int* __restrict__ dst,
                            const float* __restrict__ als, const float* __restrict__ ald,
                            float* __restrict__ ebuf, unsigned* __restrict__ menc, int ne) {
  int e = blockIdx.x * blockDim.x + threadIdx.x;
  if (e >= ne) return;
  int s = src[e], d = dst[e];
  float4 as = ((const float4*)als)[s];
  float4 ad = ((const float4*)ald)[d];
  float4 ev;
  ev.x = as.x + ad.x; ev.y = as.y + ad.y; ev.z = as.z + ad.z; ev.w = as.w + ad.w;
  ev.x = ev.x > 0.f ? ev.x : 0.2f * ev.x;
  ev.y = ev.y > 0.f ? ev.y : 0.2f * ev.y;
  ev.z = ev.z > 0.f ? ev.z : 0.2f * ev.z;
  ev.w = ev.w > 0.f ? ev.w : 0.2f * ev.w;
  ((float4*)ebuf)[e] = ev;
  atomicMax(&menc[d * 4 + 0], fenc(ev.x));
  atomicMax(&menc[d * 4 + 1], fenc(ev.y));
  atomicMax(&menc[d * 4 + 2], fenc(ev.z));
  atomicMax(&menc[d * 4 + 3], fenc(ev.w));
}

// pass B: ex = exp(e - m[dst]); store; segment sum denom
__global__ void edge_pass_b(const int* __restrict__ dst, float* __restrict__ ebuf,
                            const unsigned* __restrict__ menc, float* __restrict__ denom,
                            int ne) {
  int e = blockIdx.x * blockDim.x + threadIdx.x;
  if (e >= ne) return;
  int d = dst[e];
  float4 ev = ((const float4*)ebuf)[e];
  uint4 mu = ((const uint4*)menc)[d];
  float4 ex;
  ex.x = expf(ev.x - fdec(mu.x));
  ex.y = expf(ev.y - fdec(mu.y));
  ex.z = expf(ev.z - fdec(mu.z));
  ex.w = expf(ev.w - fdec(mu.w));
  ((float4*)ebuf)[e] = ex;
  atomicAdd(&denom[d * 4 + 0], ex.x);
  atomicAdd(&denom[d * 4 + 1], ex.y);
  atomicAdd(&denom[d * 4 + 2], ex.z);
  atomicAdd(&denom[d * 4 + 3], ex.w);
}

// pass C: agg[dst,c] += 0.25 * sum_h alpha_h * H[src, h*64+c] ; half-wave per edge
__global__ __launch_bounds__(256) void edge_pass_c(const int* __restrict__ src,
                                                   const int* __restrict__ dst,
                                                   const float* __restrict__ ebuf,
                                                   const float* __restrict__ denom,
                                                   const float* __restrict__ H,
                                                   float* __restrict__ agg, int ne) {
  int e = blockIdx.x * 16 + (threadIdx.x >> 4);  // 16 edges per block
  if (e >= ne) return;
  int l = threadIdx.x & 15;
  int s = src[e], d = dst[e];
  float4 ex = ((const float4*)ebuf)[e];
  float4 dn = ((const float4*)denom)[d];
  float a0 = ex.x / fmaxf(dn.x, 1e-16f);
  float a1 = ex.y / fmaxf(dn.y, 1e-16f);
  float a2 = ex.z / fmaxf(dn.z, 1e-16f);
  float a3 = ex.w / fmaxf(dn.w, 1e-16f);
  int c = l * 4;
  const float* Hr = H + s * 256;
  float4 v0 = *(const float4*)(Hr +   0 + c);
  float4 v1 = *(const float4*)(Hr +  64 + c);
  float4 v2 = *(const float4*)(Hr + 128 + c);
  float4 v3 = *(const float4*)(Hr + 192 + c);
  float m0 = 0.25f * (a0 * v0.x + a1 * v1.x + a2 * v2.x + a3 * v3.x);
  float m1 = 0.25f * (a0 * v0.y + a1 * v1.y + a2 * v2.y + a3 * v3.y);
  float m2 = 0.25f * (a0 * v0.z + a1 * v1.z + a2 * v2.z + a3 * v3.z);
  float m3 = 0.25f * (a0 * v0.w + a1 * v1.w + a2 * v2.w + a3 * v3.w);
  float* ap = agg + d * 64 + c;
  atomicAdd(ap + 0, m0);
  atomicAdd(ap + 1, m1);
  atomicAdd(ap + 2, m2);
  atomicAdd(ap + 3, m3);
}

extern "C" void kernel_launch(void* const* d_in, const int* in_sizes, int n_in,
                              void* d_out, int out_size, void* d_ws, size_t ws_size,
                              hipStream_t stream) {
  (void)in_sizes; (void)n_in; (void)out_size; (void)ws_size;
  const float* x_a = (const float*)d_in[0];
  const float* x_b = (const float*)d_in[1];
  const int* ei[4] = {(const int*)d_in[2], (const int*)d_in[3],
                      (const int*)d_in[4], (const int*)d_in[5]};

  char* ws = (char*)d_ws;
  size_t off = 0;
  auto alloc = [&](size_t bytes) -> void* {
    void* p = ws + off;
    off = (off + bytes + 255) & ~size_t(255);
    return p;
  };
  float*    Wp    = (float*)alloc(sizeof(float) * WP_FLOATS);
  float*    Hs    = (float*)alloc(sizeof(float) * NN * 256);
  float*    Hd    = (float*)alloc(sizeof(float) * NN * 256);
  float*    al_s  = (float*)alloc(sizeof(float) * NN * 4);
  float*    al_d  = (float*)alloc(sizeof(float) * NN * 4);
  unsigned* menc  = (unsigned*)alloc(sizeof(unsigned) * NN * 4);
  float*    denom = (float*)alloc(sizeof(float) * NN * 4);
  float*    ebuf  = (float*)alloc(sizeof(float) * EE * 4);
  float*    agg_a = (float*)alloc(sizeof(float) * NN * 64);
  float*    agg_b = (float*)alloc(sizeof(float) * NN * 64);
  float*    xa1   = (float*)alloc(sizeof(float) * NN * 64);
  float*    xb1   = (float*)alloc(sizeof(float) * NN * 64);

  const int gemm_blocks = (NN / 16 + 3) / 4;   // 782 (4 m-tiles per block)
  const int proj_blocks = (NN / 16 + 7) / 8;   // 391
  const int al_blocks   = (NN + 7) / 8;        // 6250
  const int ec_blocks   = (EE + 15) / 16;      // 31250

  for (int L = 0; L < 2; ++L) {
    int base = 6 + L * 24;
    const float* xc[2] = { L == 0 ? x_a : xa1, L == 0 ? x_b : xb1 };
    float* xo[2] = { L == 0 ? xa1 : (float*)d_out,
                     L == 0 ? xb1 : (float*)d_out + (size_t)NN * 64 };
    // agg_a = bias_aa + bias_ba ; agg_b = bias_ab + bias_bb
    init_agg<<<(NN * 64 + 255) / 256, 256, 0, stream>>>(
        agg_a, (const float*)d_in[base + 3], (const float*)d_in[base + 11], NN * 64);
    init_agg<<<(NN * 64 + 255) / 256, 256, 0, stream>>>(
        agg_b, (const float*)d_in[base + 7], (const float*)d_in[base + 15], NN * 64);

    for (int ci = 0; ci < 4; ++ci) {  // aa, ab, ba, bb
      int si = ci >> 1, ti = ci & 1;
      const float* W    = (const float*)d_in[base + ci * 4 + 0];
      const float* asrc = (const float*)d_in[base + ci * 4 + 1];
      const float* adst = (const float*)d_in[base + ci * 4 + 2];
      repack_w<<<64, 256, 0, stream>>>(W, Wp);
      gemm_64x256<<<gemm_blocks, 256, 0, stream>>>(xc[si], Wp, Hs, NN);
      const float* Hdp = Hs;
      if (si != ti) {
        gemm_64x256<<<gemm_blocks, 256, 0, stream>>>(xc[ti], Wp, Hd, NN);
        Hdp = Hd;
      }
      attn_logits<<<al_blocks, 256, 0, stream>>>(Hs, asrc, al_s, NN);
      attn_logits<<<al_blocks, 256, 0, stream>>>(Hdp, adst, al_d, NN);
      init_md<<<(NN * 4 + 255) / 256, 256, 0, stream>>>(menc, denom, NN * 4);
      const int* esrc = ei[ci];
      const int* edst = ei[ci] + EE;
      edge_pass_a<<<(EE + 255) / 256, 256, 0, stream>>>(esrc, edst, al_s, al_d, ebuf, menc, EE);
      edge_pass_b<<<(EE + 255) / 256, 256, 0, stream>>>(edst, ebuf, menc, denom, EE);
      edge_pass_c<<<ec_blocks, 256, 0, stream>>>(esrc, edst, ebuf, denom, Hs,
                                                 ti == 0 ? agg_a : agg_b, EE);
    }
    // x_out = relu(agg) @ projW + projb + x @ skipW + skipb
    out_proj<<<proj_blocks, 256, 0, stream>>>(
        agg_a, xc[0], (const float*)d_in[base + 16], (const float*)d_in[base + 20],
        (const float*)d_in[base + 17], (const float*)d_in[base + 21], xo[0], NN);
    out_proj<<<proj_blocks, 256, 0, stream>>>(
        agg_b, xc[1], (const float*)d_in[base + 18], (const float*)d_in[base + 22],
        (const float*)d_in[base + 19], (const float*)d_in[base + 23], xo[1], NN);
  }
}